// SVDHead_68667937129014
// MI455X (gfx1250) — compile-verified
//
#include <hip/hip_runtime.h>
#include <math.h>

// Problem constants from the reference.
#define BATCH    512
#define NPTS     256
#define SENT     256
#define ROWS     16          // score rows per workgroup (one WMMA M-tile)
#define EPITCH   260         // exp-tile row pitch (floats); 4m+k bank pattern is conflict-free

typedef __attribute__((ext_vector_type(2))) float v2f;
typedef __attribute__((ext_vector_type(8))) float v8f;

__device__ __forceinline__ float wave_sum(float v) {
#pragma unroll
  for (int off = 16; off > 0; off >>= 1) v += __shfl_xor(v, off, 32);
  return v;
}
__device__ __forceinline__ float wave_max(float v) {
#pragma unroll
  for (int off = 16; off > 0; off >>= 1) v = fmaxf(v, __shfl_xor(v, off, 32));
  return v;
}

// ---------------------------------------------------------------------------
// Kernel 1: fused row-softmax + two dense (16x256)x(256x16-padded) WMMA chains.
// grid = (NPTS/ROWS, BATCH), block = 256 threads (8 waves).
// Gather trick: c[p,d] = sum_j m_j * E[p,j] * kpts1[j,d] with multiplicities
// m_j = #{q : valid[q] && mc[q]==j}  -> both matmuls are dense and share A.
// Wave 0 runs both WMMA chains (accF: c_f, accG: c) off one A fragment.
// ---------------------------------------------------------------------------
__global__ __launch_bounds__(256) void svdhead_softmax_wmma(
    const float* __restrict__ scores,   // (B, N, N)
    const float* __restrict__ tgt,      // (B, 3, N)
    const int*   __restrict__ indices0, // (B, N)
    float* __restrict__ cF,             // (B, N, 3) full-branch correspondence
    float* __restrict__ cW)             // (B, N, 3) valid-branch correspondence
{
  __shared__ float Etile[ROWS * EPITCH]; // un-normalized exp(scores - rowmax)
  __shared__ float Bfull[NPTS * 16];     // kpts1 rows, cols 3..15 zero
  __shared__ float Bgat [NPTS * 16];     // m_j * kpts1 rows, cols 3..15 zero
  __shared__ int   cntS [NPTS];          // gather multiplicities m_j
  __shared__ float rdenS[ROWS];          // reciprocal softmax denominators

  const int b    = blockIdx.y;
  const int rb   = blockIdx.x;
  const int tid  = threadIdx.x;
  const int lane = tid & 31;
  const int wave = tid >> 5;

  // ---- multiplicity histogram + dense B tile (1 thread per index q) ----
  const float* tb = tgt + (size_t)b * 3 * NPTS;
  {
    const int q = tid;                                // exactly 256 threads
    cntS[q] = 0;
#pragma unroll
    for (int d = 0; d < 16; ++d)
      Bfull[q * 16 + d] = (d < 3) ? tb[d * NPTS + q] : 0.0f;
  }
  __syncthreads();
  {
    const int q = tid;
    const int idx = indices0[(size_t)b * NPTS + q];
    if (idx != SENT) {
      int mcq = idx < 0 ? 0 : (idx > NPTS - 1 ? NPTS - 1 : idx);
      atomicAdd(&cntS[mcq], 1);
    }
  }

  // ---- fused softmax (no dependency on cntS): wave owns rows {w, w+8} ----
  for (int r = wave; r < ROWS; r += 8) {
    const float4* srow4 = reinterpret_cast<const float4*>(
        scores + ((size_t)b * NPTS + (size_t)rb * ROWS + r) * NPTS);
    const float4 x0 = srow4[lane * 2 + 0];
    const float4 x1 = srow4[lane * 2 + 1];
    float m = fmaxf(fmaxf(fmaxf(x0.x, x0.y), fmaxf(x0.z, x0.w)),
                    fmaxf(fmaxf(x1.x, x1.y), fmaxf(x1.z, x1.w)));
    m = wave_max(m);
    float4 e0, e1;
    e0.x = __expf(x0.x - m); e0.y = __expf(x0.y - m);
    e0.z = __expf(x0.z - m); e0.w = __expf(x0.w - m);
    e1.x = __expf(x1.x - m); e1.y = __expf(x1.y - m);
    e1.z = __expf(x1.z - m); e1.w = __expf(x1.w - m);
    float s = (e0.x + e0.y + e0.z + e0.w) + (e1.x + e1.y + e1.z + e1.w);
    s = wave_sum(s);
    float4* erow4 = reinterpret_cast<float4*>(&Etile[r * EPITCH]);
    erow4[lane * 2 + 0] = e0;
    erow4[lane * 2 + 1] = e1;
    if (lane == 0) rdenS[r] = 1.0f / s;
  }
  __syncthreads();

  // ---- weighted B tile for the gathered branch (needs final cntS) ----
  {
    const int q = tid;
    const float w = (float)cntS[q];
#pragma unroll
    for (int d = 0; d < 16; ++d)
      Bgat[q * 16 + d] = (d < 3) ? w * tb[d * NPTS + q] : 0.0f;
  }
  __syncthreads();

  // ---- dual WMMA chain: K = 256 in steps of 4 (V_WMMA_F32_16X16X4_F32) ----
  if (wave == 0) {
    v8f accF = (v8f){0.f, 0.f, 0.f, 0.f, 0.f, 0.f, 0.f, 0.f};
    v8f accG = (v8f){0.f, 0.f, 0.f, 0.f, 0.f, 0.f, 0.f, 0.f};
    const int m  = lane & 15;            // A: row M ; B/C: col N
    const int kk = (lane >> 4) * 2;      // K sub-offset per ISA layout
    const float* Erow = Etile + m * EPITCH;
#pragma unroll 4
    for (int k = 0; k < NPTS; k += 4) {
      const int k0 = k + kk;
      v2f a, bf, bg;
      a.x  = Erow[k0];
      a.y  = Erow[k0 + 1];
      bf.x = Bfull[k0 * 16 + m];
      bf.y = Bfull[(k0 + 1) * 16 + m];
      bg.x = Bgat[k0 * 16 + m];
      bg.y = Bgat[(k0 + 1) * 16 + m];
      accF = __builtin_amdgcn_wmma_f32_16x16x4_f32(
                 false, a, false, bf, (short)0, accF, false, false);
      accG = __builtin_amdgcn_wmma_f32_16x16x4_f32(
                 false, a, false, bg, (short)0, accG, false, false);
    }
    // C/D layout: VGPR r -> M = r + 8*(lane>>4), N = lane&15
    const int n = lane & 15;
    if (n < 3) {
#pragma unroll
      for (int r = 0; r < 8; ++r) {
        const int M = r + (lane >> 4) * 8;
        const int p = rb * ROWS + M;
        const float rd = rdenS[M];
        cF[((size_t)b * NPTS + p) * 3 + n] = accF[r] * rd;
        cW[((size_t)b * NPTS + p) * 3 + n] = accG[r] * rd;
      }
    }
  }
}

// ---------------------------------------------------------------------------
// Kernel 2: per-batch reductions + 3x3 Kabsch (Jacobi eigen of H^T H).
// grid = BATCH, block = 32 (one wave).
// ---------------------------------------------------------------------------
__global__ __launch_bounds__(32) void svdhead_procrustes(
    const float* __restrict__ src,      // (B, 3, N)
    const int*   __restrict__ indices0, // (B, N)
    const float* __restrict__ cW,       // (B, N, 3)
    const float* __restrict__ cF,       // (B, N, 3)
    float* __restrict__ outR,           // (B, 3, 3)
    float* __restrict__ outT)           // (B, 3)
{
  const int b    = blockIdx.x;
  const int lane = threadIdx.x;

  float cnt = 0.f;
  float vsx[3] = {0,0,0}, vsy[3] = {0,0,0}, vsxy[9] = {0,0,0,0,0,0,0,0,0};
  float fsx[3] = {0,0,0}, fsy[3] = {0,0,0}, fsxy[9] = {0,0,0,0,0,0,0,0,0};

  const float* sb = src + (size_t)b * 3 * NPTS;
  for (int n = lane; n < NPTS; n += 32) {
    const float x[3] = { sb[0 * NPTS + n], sb[1 * NPTS + n], sb[2 * NPTS + n] };
    const float* cf = cF + ((size_t)b * NPTS + n) * 3;
    const float yf[3] = { cf[0], cf[1], cf[2] };
#pragma unroll
    for (int d = 0; d < 3; ++d) { fsx[d] += x[d]; fsy[d] += yf[d]; }
#pragma unroll
    for (int d = 0; d < 3; ++d)
#pragma unroll
      for (int e = 0; e < 3; ++e) fsxy[d * 3 + e] += x[d] * yf[e];

    if (indices0[(size_t)b * NPTS + n] != SENT) {
      const float* cw = cW + ((size_t)b * NPTS + n) * 3;
      const float yv[3] = { cw[0], cw[1], cw[2] };
      cnt += 1.f;
#pragma unroll
      for (int d = 0; d < 3; ++d) { vsx[d] += x[d]; vsy[d] += yv[d]; }
#pragma unroll
      for (int d = 0; d < 3; ++d)
#pragma unroll
        for (int e = 0; e < 3; ++e) vsxy[d * 3 + e] += x[d] * yv[e];
    }
  }

  cnt = wave_sum(cnt);
#pragma unroll
  for (int d = 0; d < 3; ++d) {
    vsx[d] = wave_sum(vsx[d]); vsy[d] = wave_sum(vsy[d]);
    fsx[d] = wave_sum(fsx[d]); fsy[d] = wave_sum(fsy[d]);
  }
#pragma unroll
  for (int k = 0; k < 9; ++k) { vsxy[k] = wave_sum(vsxy[k]); fsxy[k] = wave_sum(fsxy[k]); }

  if (lane != 0) return;

  // Select branch: valid if M > 3 else full.  H = Sxy - Sx*Sy^T / M.
  const bool use_v = cnt > 3.0f;
  float mu[3], H[3][3];
  if (use_v) {
    const float inv = 1.0f / fmaxf(cnt, 1.0f);
#pragma unroll
    for (int d = 0; d < 3; ++d) mu[d] = vsx[d] * inv;
#pragma unroll
    for (int d = 0; d < 3; ++d)
#pragma unroll
      for (int e = 0; e < 3; ++e) H[d][e] = vsxy[d * 3 + e] - vsx[d] * vsy[e] * inv;
  } else {
    const float inv = 1.0f / (float)NPTS;
#pragma unroll
    for (int d = 0; d < 3; ++d) mu[d] = fsx[d] * inv;
#pragma unroll
    for (int d = 0; d < 3; ++d)
#pragma unroll
      for (int e = 0; e < 3; ++e) H[d][e] = fsxy[d * 3 + e] - fsx[d] * fsy[e] * inv;
  }

  // A = H^T H, Jacobi eigen -> right singular vectors V (det(V)=+1).
  float A[3][3], V[3][3] = {{1,0,0},{0,1,0},{0,0,1}};
#pragma unroll
  for (int i = 0; i < 3; ++i)
#pragma unroll
    for (int j = 0; j < 3; ++j)
      A[i][j] = H[0][i] * H[0][j] + H[1][i] * H[1][j] + H[2][i] * H[2][j];

  for (int sweep = 0; sweep < 12; ++sweep) {
#pragma unroll
    for (int pi = 0; pi < 3; ++pi) {
      const int p = (pi == 2) ? 1 : 0;
      const int q = (pi == 0) ? 1 : 2;
      const float apq = A[p][q];
      if (fabsf(apq) > 1e-30f) {
        const float tau = (A[q][q] - A[p][p]) / (2.0f * apq);
        const float t   = (tau >= 0.f ? 1.f : -1.f) / (fabsf(tau) + sqrtf(1.f + tau * tau));
        const float c   = 1.0f / sqrtf(1.f + t * t);
        const float s   = t * c;
#pragma unroll
        for (int k = 0; k < 3; ++k) {
          const float akp = A[k][p], akq = A[k][q];
          A[k][p] = c * akp - s * akq;  A[k][q] = s * akp + c * akq;
        }
#pragma unroll
        for (int k = 0; k < 3; ++k) {
          const float apk = A[p][k], aqk = A[q][k];
          A[p][k] = c * apk - s * aqk;  A[q][k] = s * apk + c * aqk;
        }
#pragma unroll
        for (int k = 0; k < 3; ++k) {
          const float vkp = V[k][p], vkq = V[k][q];
          V[k][p] = c * vkp - s * vkq;  V[k][q] = s * vkp + c * vkq;
        }
      }
    }
  }

  // Sort eigenvalues descending (track column order).
  float lam[3] = { A[0][0], A[1][1], A[2][2] };
  int id[3] = { 0, 1, 2 };
#pragma unroll
  for (int i = 0; i < 2; ++i)
#pragma unroll
    for (int j = 0; j < 2 - i; ++j)
      if (lam[id[j]] < lam[id[j + 1]]) { int tmp = id[j]; id[j] = id[j + 1]; id[j + 1] = tmp; }

  float v0[3], v1[3], v2[3];
#pragma unroll
  for (int d = 0; d < 3; ++d) { v0[d] = V[d][id[0]]; v1[d] = V[d][id[1]]; v2[d] = V[d][id[2]]; }
  const float s0 = sqrtf(fmaxf(lam[id[0]], 0.f));
  const float s1 = sqrtf(fmaxf(lam[id[1]], 0.f));

  float R[3][3], u0[3], u1[3], u2[3];
  if (s0 < 1e-20f) {
    // H ~ 0: identity rotation.
#pragma unroll
    for (int i = 0; i < 3; ++i)
#pragma unroll
      for (int j = 0; j < 3; ++j) R[i][j] = (i == j) ? 1.f : 0.f;
  } else {
#pragma unroll
    for (int d = 0; d < 3; ++d) {
      u0[d] = (H[d][0] * v0[0] + H[d][1] * v0[1] + H[d][2] * v0[2]) / s0;
      u1[d] = (H[d][0] * v1[0] + H[d][1] * v1[1] + H[d][2] * v1[2]) / fmaxf(s1, 1e-20f);
    }
    // Re-orthonormalize u1 against u0 (robust for small s1).
    const float d01 = u0[0] * u1[0] + u0[1] * u1[1] + u0[2] * u1[2];
#pragma unroll
    for (int d = 0; d < 3; ++d) u1[d] -= d01 * u0[d];
    float n1 = sqrtf(u1[0] * u1[0] + u1[1] * u1[1] + u1[2] * u1[2]);
    if (n1 < 1e-12f) { // pick any unit vector orthogonal to u0
      u1[0] = -u0[1]; u1[1] = u0[0]; u1[2] = 0.f;
      n1 = sqrtf(u1[0] * u1[0] + u1[1] * u1[1] + u1[2] * u1[2]);
      if (n1 < 1e-12f) { u1[0] = 0.f; u1[1] = -u0[2]; u1[2] = u0[1];
        n1 = sqrtf(u1[0] * u1[0] + u1[1] * u1[1] + u1[2] * u1[2]); }
    }
#pragma unroll
    for (int d = 0; d < 3; ++d) u1[d] /= n1;
    // u2 = u0 x u1 reproduces the reference's det-based third-column flip.
    u2[0] = u0[1] * u1[2] - u0[2] * u1[1];
    u2[1] = u0[2] * u1[0] - u0[0] * u1[2];
    u2[2] = u0[0] * u1[1] - u0[1] * u1[0];
#pragma unroll
    for (int i = 0; i < 3; ++i)
#pragma unroll
      for (int j = 0; j < 3; ++j)
        R[i][j] = v0[i] * u0[j] + v1[i] * u1[j] + v2[i] * u2[j];
  }

#pragma unroll
  for (int i = 0; i < 3; ++i)
#pragma unroll
    for (int j = 0; j < 3; ++j) outR[(size_t)b * 9 + i * 3 + j] = R[i][j];
#pragma unroll
  for (int i = 0; i < 3; ++i)
    outT[(size_t)b * 3 + i] = -(R[i][0] * mu[0] + R[i][1] * mu[1] + R[i][2] * mu[2]);
}

// ---------------------------------------------------------------------------
extern "C" void kernel_launch(void* const* d_in, const int* in_sizes, int n_in,
                              void* d_out, int out_size, void* d_ws, size_t ws_size,
                              hipStream_t stream) {
  const float* src      = (const float*)d_in[0];   // (B,3,N)
  const float* tgt      = (const float*)d_in[1];   // (B,3,N)
  const float* scores   = (const float*)d_in[2];   // (B,N,N)
  const int*   indices0 = (const int*)d_in[3];     // (B,N)
  // d_in[4] (indices1) is unused by the reference forward pass.

  float* cF = (float*)d_ws;                        // (B,N,3)
  float* cW = cF + (size_t)BATCH * NPTS * 3;       // (B,N,3)  (~3 MB total)

  dim3 grid1(NPTS / ROWS, BATCH);
  svdhead_softmax_wmma<<<grid1, 256, 0, stream>>>(scores, tgt, indices0, cF, cW);

  float* outR = (float*)d_out;                     // (B,3,3) first
  float* outT = outR + (size_t)BATCH * 9;          // (B,3)   second
  svdhead_procrustes<<<BATCH, 32, 0, stream>>>(src, indices0, cW, cF, outR, outT);
}